// RGDTLayer_28441273434208
// MI455X (gfx1250) — compile-verified
//
#include <hip/hip_runtime.h>
#include <math.h>

#define NN   100000
#define EE   1600000
#define RR   100
#define HH   8
#define DHH  16
#define DD   128
#define DFF  512
#define NHOPS 4
#define ALPHA 0.1f

typedef float v8f __attribute__((ext_vector_type(8)));
typedef float v2f __attribute__((ext_vector_type(2)));

// ---------------- WMMA helper: D = A(16x4,f32) * B(4x16,f32) + C (f32 16x16) ----
static __device__ __forceinline__ v8f wmma4(v2f a, v2f b, v8f c) {
  return __builtin_amdgcn_wmma_f32_16x16x4_f32(false, a, false, b, (short)0, c,
                                               false, false);
}

static __device__ __forceinline__ float waveAllSum(float v) {
#pragma unroll
  for (int o = 16; o > 0; o >>= 1) v += __shfl_xor(v, o, 32);
  return v;
}

// native f32 global atomic add (global_atomic_add_f32), not a CAS loop
static __device__ __forceinline__ void atomAddF(float* p, float v) {
  unsafeAtomicAdd(p, v);
}

// monotonic float<->uint encoding for atomicMax over signed floats
static __device__ __forceinline__ unsigned encf(float f) {
  unsigned u = __float_as_uint(f);
  return (u & 0x80000000u) ? ~u : (u | 0x80000000u);
}
static __device__ __forceinline__ float decf(unsigned u) {
  u = (u & 0x80000000u) ? (u & 0x7FFFFFFFu) : ~u;
  return __uint_as_float(u);
}

// ---------------------------------------------------------------------------
// LayerNorm(x rows) then up to 3 projections: o_m = LN(x) @ W_m  (D x D each)
// block = 768 threads = 24 waves; one 16-row tile per block.
// waves 0..15: LN one row each into LDS. wave w: matrix m=w/8, colTile c=w%8.
// ---------------------------------------------------------------------------
#define SXS 132
__global__ __launch_bounds__(768) void ln_proj3_kernel(
    const float* __restrict__ x, int nrows, const float* __restrict__ g,
    const float* __restrict__ b, const float* __restrict__ W0,
    float* __restrict__ o0, const float* __restrict__ W1,
    float* __restrict__ o1, const float* __restrict__ W2,
    float* __restrict__ o2) {
  __shared__ float sx[16 * SXS];
  const int tid = threadIdx.x;
  const int wave = tid >> 5, lane = tid & 31;
  const int row0 = blockIdx.x * 16;

  if (wave < 16) {
    const int row = row0 + wave;
    float xv[4];
#pragma unroll
    for (int j = 0; j < 4; j++) {
      const int col = lane + 32 * j;
      xv[j] = (row < nrows) ? x[(size_t)row * DD + col] : 0.f;
    }
    float s = waveAllSum(xv[0] + xv[1] + xv[2] + xv[3]);
    const float mean = s * (1.f / DD);
    float vs = 0.f;
#pragma unroll
    for (int j = 0; j < 4; j++) {
      const float d = xv[j] - mean;
      vs += d * d;
    }
    vs = waveAllSum(vs);
    const float r = rsqrtf(vs * (1.f / DD) + 1e-5f);
#pragma unroll
    for (int j = 0; j < 4; j++) {
      const int col = lane + 32 * j;
      sx[wave * SXS + col] = (xv[j] - mean) * r * g[col] + b[col];
    }
  }
  __syncthreads();

  const int m = wave >> 3, c = wave & 7;
  const float* W = (m == 0) ? W0 : ((m == 1) ? W1 : W2);
  float* out = (m == 0) ? o0 : ((m == 1) ? o1 : o2);
  if (out == nullptr) return;

  const int half = lane >> 4;   // 0: K lo pair, 1: K hi pair
  const int ln15 = lane & 15;
  const int col = c * 16 + ln15;
  v8f acc = {};
  for (int kb = 0; kb < DD; kb += 4) {
    const int ka = kb + (half << 1);
    v2f A = {sx[ln15 * SXS + ka], sx[ln15 * SXS + ka + 1]};
    v2f B = {W[(size_t)ka * DD + col], W[(size_t)(ka + 1) * DD + col]};
    acc = wmma4(A, B, acc);
  }
#pragma unroll
  for (int i = 0; i < 8; i++) {
    const int row = row0 + i + 8 * half;
    if (row < nrows) out[(size_t)row * DD + col] = acc[i];
  }
}

// ---------------------------------------------------------------------------
// Fused FFN: out = relu(LN(rst)@W1+b1)@W2 + b2 + rst  (16-row tile per block)
// ---------------------------------------------------------------------------
#define SYS 132
#define SHS 516
__global__ __launch_bounds__(1024) void ffn_kernel(
    const float* __restrict__ rst, const float* __restrict__ g,
    const float* __restrict__ b, const float* __restrict__ W1,
    const float* __restrict__ b1, const float* __restrict__ W2,
    const float* __restrict__ b2, float* __restrict__ out) {
  __shared__ float sy[16 * SYS];
  __shared__ float sh[16 * SHS];
  const int tid = threadIdx.x;
  const int wave = tid >> 5, lane = tid & 31;
  const int row0 = blockIdx.x * 16;

  if (wave < 16) {
    const int row = row0 + wave;
    float xv[4];
#pragma unroll
    for (int j = 0; j < 4; j++) {
      const int col = lane + 32 * j;
      xv[j] = (row < NN) ? rst[(size_t)row * DD + col] : 0.f;
    }
    float s = waveAllSum(xv[0] + xv[1] + xv[2] + xv[3]);
    const float mean = s * (1.f / DD);
    float vs = 0.f;
#pragma unroll
    for (int j = 0; j < 4; j++) {
      const float d = xv[j] - mean;
      vs += d * d;
    }
    vs = waveAllSum(vs);
    const float r = rsqrtf(vs * (1.f / DD) + 1e-5f);
#pragma unroll
    for (int j = 0; j < 4; j++) {
      const int col = lane + 32 * j;
      sy[wave * SYS + col] = (xv[j] - mean) * r * g[col] + b[col];
    }
  }
  __syncthreads();

  const int half = lane >> 4;
  const int ln15 = lane & 15;
  {  // hidden = relu(y@W1 + b1): 32 waves x one 16-col tile of DFF=512
    const int col = wave * 16 + ln15;
    v8f acc = {};
    for (int kb = 0; kb < DD; kb += 4) {
      const int ka = kb + (half << 1);
      v2f A = {sy[ln15 * SYS + ka], sy[ln15 * SYS + ka + 1]};
      v2f B = {W1[(size_t)ka * DFF + col], W1[(size_t)(ka + 1) * DFF + col]};
      acc = wmma4(A, B, acc);
    }
    const float bias = b1[col];
#pragma unroll
    for (int i = 0; i < 8; i++) {
      const int M = i + 8 * half;
      const float hv = acc[i] + bias;
      sh[M * SHS + col] = hv > 0.f ? hv : 0.f;
    }
  }
  __syncthreads();

  if (wave < 8) {  // out = hidden@W2 + b2 + rst
    const int col = wave * 16 + ln15;
    v8f acc = {};
    for (int kb = 0; kb < DFF; kb += 4) {
      const int ka = kb + (half << 1);
      v2f A = {sh[ln15 * SHS + ka], sh[ln15 * SHS + ka + 1]};
      v2f B = {W2[(size_t)ka * DD + col], W2[(size_t)(ka + 1) * DD + col]};
      acc = wmma4(A, B, acc);
    }
    const float bias = b2[col];
#pragma unroll
    for (int i = 0; i < 8; i++) {
      const int row = row0 + i + 8 * half;
      if (row < NN)
        out[(size_t)row * DD + col] = acc[i] + bias + rst[(size_t)row * DD + col];
    }
  }
}

// ------------------------------ graph kernels ------------------------------
__global__ void fill_kernel(float* __restrict__ p, size_t n, float v) {
  size_t i = (size_t)blockIdx.x * blockDim.x + threadIdx.x;
  const size_t stride = (size_t)gridDim.x * blockDim.x;
  for (; i < n; i += stride) p[i] = v;
}

__global__ void count_deg_kernel(const int* __restrict__ dst,
                                 float* __restrict__ deg) {
  const int e = blockIdx.x * blockDim.x + threadIdx.x;
  if (e < EE) atomAddF(&deg[dst[e]], 1.0f);
}

__global__ void log1p_kernel(float* __restrict__ p) {
  const int i = blockIdx.x * blockDim.x + threadIdx.x;
  if (i < NN) p[i] = log1pf(p[i]);
}

__global__ void edge_logit_kernel(const float* __restrict__ k,
                                  const float* __restrict__ q,
                                  const float* __restrict__ rel,
                                  const int* __restrict__ src,
                                  const int* __restrict__ dst,
                                  const int* __restrict__ rid,
                                  const float* __restrict__ log_in,
                                  float* __restrict__ e_out,
                                  unsigned* __restrict__ emaxU) {
  const int idx = blockIdx.x * blockDim.x + threadIdx.x;
  if (idx >= EE * HH) return;
  const int e = idx >> 3, h = idx & 7;
  const int s = src[e], d = dst[e], r = rid[e];
  const float4* kp = (const float4*)(k + (size_t)s * DD + h * DHH);
  const float4* qp = (const float4*)(q + (size_t)d * DD + h * DHH);
  const float4* rp = (const float4*)(rel + (size_t)r * DD + h * DHH);
  float dot = 0.f;
#pragma unroll
  for (int j = 0; j < 4; j++) {
    const float4 a = kp[j], bq = qp[j], rr = rp[j];
    dot += a.x * rr.x * bq.x + a.y * rr.y * bq.y + a.z * rr.z * bq.z +
           a.w * rr.w * bq.w;
  }
  const float val = dot * log_in[d] * 0.25f;  // 1/sqrt(DH)
  e_out[idx] = val;
  atomicMax(&emaxU[d * HH + h], encf(val));
}

__global__ void edge_exp_kernel(float* __restrict__ a,
                                const int* __restrict__ dst,
                                const unsigned* __restrict__ emaxU,
                                float* __restrict__ denom) {
  const int idx = blockIdx.x * blockDim.x + threadIdx.x;
  if (idx >= EE * HH) return;
  const int e = idx >> 3, h = idx & 7;
  const int d = dst[e];
  const float ex = expf(a[idx] - decf(emaxU[d * HH + h]));
  a[idx] = ex;
  atomAddF(&denom[d * HH + h], ex);
}

__global__ void edge_norm_kernel(float* __restrict__ a,
                                 const int* __restrict__ dst,
                                 const float* __restrict__ denom) {
  const int idx = blockIdx.x * blockDim.x + threadIdx.x;
  if (idx >= EE * HH) return;
  const int e = idx >> 3, h = idx & 7;
  a[idx] /= denom[dst[e] * HH + h];
}

__global__ void scatter_kernel(const float* __restrict__ a,
                               const int* __restrict__ src,
                               const int* __restrict__ dst,
                               const float* __restrict__ fin,
                               float* __restrict__ fout) {
  const int idx = blockIdx.x * blockDim.x + threadIdx.x;
  if (idx >= EE * HH) return;
  const int e = idx >> 3, h = idx & 7;
  const float w = a[idx];
  const float4* fi = (const float4*)(fin + (size_t)src[e] * DD + h * DHH);
  float* fo = fout + (size_t)dst[e] * DD + h * DHH;
  float4 m[4];
#pragma unroll
  for (int j = 0; j < 4; j++) {
    const float4 x = fi[j];
    m[j].x = x.x * w; m[j].y = x.y * w; m[j].z = x.z * w; m[j].w = x.w * w;
  }
#pragma unroll
  for (int j = 0; j < 4; j++) {
    atomAddF(&fo[4 * j + 0], m[j].x);
    atomAddF(&fo[4 * j + 1], m[j].y);
    atomAddF(&fo[4 * j + 2], m[j].z);
    atomAddF(&fo[4 * j + 3], m[j].w);
  }
}

__global__ void blend_kernel(float* __restrict__ f,
                             const float* __restrict__ f0) {
  const int i = blockIdx.x * blockDim.x + threadIdx.x;
  if (i < NN * DD) f[i] = (1.0f - ALPHA) * f[i] + ALPHA * f0[i];
}

__global__ void residual_kernel(const float* __restrict__ feat,
                                const float* __restrict__ ent,
                                float* __restrict__ rst) {
  const int i = blockIdx.x * blockDim.x + threadIdx.x;
  if (i < NN * DD) rst[i] = feat[i] + ent[i];
}

// ---------------------------------------------------------------------------
extern "C" void kernel_launch(void* const* d_in, const int* in_sizes, int n_in,
                              void* d_out, int out_size, void* d_ws,
                              size_t ws_size, hipStream_t stream) {
  const float* ent = (const float*)d_in[0];
  const float* relf = (const float*)d_in[1];
  const float* W_head = (const float*)d_in[2];
  const float* W_tail = (const float*)d_in[3];
  const float* W_ent = (const float*)d_in[4];
  const float* W_rel = (const float*)d_in[5];
  const float* g_ent = (const float*)d_in[6];
  const float* b_ent = (const float*)d_in[7];
  const float* g_rel = (const float*)d_in[8];
  const float* b_rel = (const float*)d_in[9];
  const float* g_ff = (const float*)d_in[10];
  const float* b_ff = (const float*)d_in[11];
  const float* W1 = (const float*)d_in[12];
  const float* b1 = (const float*)d_in[13];
  const float* W2 = (const float*)d_in[14];
  const float* b2 = (const float*)d_in[15];
  const int* src = (const int*)d_in[16];
  const int* dst = (const int*)d_in[17];
  const int* rid = (const int*)d_in[18];
  float* out = (float*)d_out;

  float* ws = (float*)d_ws;
  size_t off = 0;
  float* kbuf = ws + off;  off += (size_t)NN * DD;
  float* qbuf = ws + off;  off += (size_t)NN * DD;
  float* vbuf = ws + off;  off += (size_t)NN * DD;
  float* relp = ws + off;  off += (size_t)112 * DD;
  float* abuf = ws + off;  off += (size_t)EE * HH;
  float* login = ws + off; off += (size_t)NN;
  unsigned* emaxU = (unsigned*)(ws + off); off += (size_t)NN * HH;
  float* denom = ws + off; off += (size_t)NN * HH;

  const int EHT = EE * HH;
  const int eb = (EHT + 255) / 256;
  const int ndb = (NN * DD + 255) / 256;

  // zero stats region (login + emax(encoded-min==0) + denom)
  fill_kernel<<<2048, 256, 0, stream>>>(login, (size_t)NN + 2 * (size_t)NN * HH,
                                        0.f);
  // k,q,v projections + rel projection (WMMA f32)
  ln_proj3_kernel<<<(NN + 15) / 16, 768, 0, stream>>>(
      ent, NN, g_ent, b_ent, W_head, kbuf, W_tail, qbuf, W_ent, vbuf);
  ln_proj3_kernel<<<(RR + 15) / 16, 768, 0, stream>>>(
      relf, RR, g_rel, b_rel, W_rel, relp, W_rel, nullptr, W_rel, nullptr);

  // degree -> log1p
  count_deg_kernel<<<(EE + 255) / 256, 256, 0, stream>>>(dst, login);
  log1p_kernel<<<(NN + 255) / 256, 256, 0, stream>>>(login);

  // attention logits + segment softmax
  edge_logit_kernel<<<eb, 256, 0, stream>>>(kbuf, qbuf, relp, src, dst, rid,
                                            login, abuf, emaxU);
  edge_exp_kernel<<<eb, 256, 0, stream>>>(abuf, dst, emaxU, denom);
  edge_norm_kernel<<<eb, 256, 0, stream>>>(abuf, dst, denom);

  // PPR diffusion hops (feat0 = vbuf; ping-pong kbuf/qbuf)
  float* fin = vbuf;
  float* fout = kbuf;
  for (int hop = 0; hop < NHOPS; hop++) {
    fill_kernel<<<2048, 256, 0, stream>>>(fout, (size_t)NN * DD, 0.f);
    scatter_kernel<<<eb, 256, 0, stream>>>(abuf, src, dst, fin, fout);
    blend_kernel<<<ndb, 256, 0, stream>>>(fout, vbuf);
    fin = fout;
    fout = (fout == kbuf) ? qbuf : kbuf;
  }
  // final feat is in `fin`; rst = feat + ent_feat -> vbuf (feat0 now free)
  residual_kernel<<<ndb, 256, 0, stream>>>(fin, ent, vbuf);

  // fused FFN + residual -> d_out (WMMA f32)
  ffn_kernel<<<(NN + 15) / 16, 1024, 0, stream>>>(vbuf, g_ff, b_ff, W1, b1, W2,
                                                  b2, out);
}